// Wasserstein1d_40303973105804
// MI455X (gfx1250) — compile-verified
//
#include <hip/hip_runtime.h>

// Shapes fixed by the reference setup_inputs(): (nb, nt, nr, nc) = (4, 2048, 64, 32)
#define NB   4
#define NT   2048
#define NR   64
#define NC   32
#define NPIX (NR * NC)            // 2048 pixels per batch (contiguous fastest dims)
#define NTRACE (NB * NPIX)        // 8192 traces
#define TPB  256
#define EPT  (NT / TPB)           // 8 elements per thread
#define DTF  0.001f

typedef __attribute__((ext_vector_type(2))) float v2f;
typedef __attribute__((ext_vector_type(8))) float v8f;

// ---------------- Kernel 1: per-block partial min over x and y ----------------
__global__ __launch_bounds__(TPB)
void kmin_partial(const float* __restrict__ x, const float* __restrict__ y,
                  float* __restrict__ part, int n) {
    __shared__ float red[TPB];
    int tid = threadIdx.x;
    float m = 3.4028235e38f;
    int stride = gridDim.x * blockDim.x;
    for (int i = blockIdx.x * blockDim.x + tid; i < n; i += stride) {
        m = fminf(m, x[i]);
        m = fminf(m, y[i]);
    }
    red[tid] = m;
    __syncthreads();
    for (int s = TPB / 2; s > 0; s >>= 1) {
        if (tid < s) red[tid] = fminf(red[tid], red[tid + s]);
        __syncthreads();
    }
    if (tid == 0) part[blockIdx.x] = red[0];
}

// ---------------- Kernel 2: final min over the partials ----------------
__global__ __launch_bounds__(TPB)
void kmin_final(const float* __restrict__ part, float* __restrict__ mn_out, int nparts) {
    __shared__ float red[TPB];
    int tid = threadIdx.x;
    float m = 3.4028235e38f;
    for (int i = tid; i < nparts; i += TPB) m = fminf(m, part[i]);
    red[tid] = m;
    __syncthreads();
    for (int s = TPB / 2; s > 0; s >>= 1) {
        if (tid < s) red[tid] = fminf(red[tid], red[tid + s]);
        __syncthreads();
    }
    if (tid == 0) mn_out[0] = red[0];
}

// Meta binary search: number of elements < q in sorted a[0..2047] (== searchsorted left)
__device__ __forceinline__ int lower_bound_nt(const float* a, float q) {
    int lo = 0;
    #pragma unroll
    for (int step = NT / 2; step > 0; step >>= 1) {
        int m = lo + step;
        if (m <= NT && a[m - 1] < q) lo = m;
    }
    return lo;
}

// Hierarchical inclusive scan of TPB*EPT values held EPT-per-thread; writes the
// inclusive cumsum normalized by the total into lds_out; returns the total.
__device__ __forceinline__ float scan_normalize(float* lds_out, const float local[EPT],
                                                int tid, float* wsumbuf) {
    float incl[EPT];
    float run = 0.0f;
    #pragma unroll
    for (int k = 0; k < EPT; ++k) { run += local[k]; incl[k] = run; }
    float tsum = run;

    int lane = tid & 31, wid = tid >> 5;
    float v = tsum;
    #pragma unroll
    for (int d = 1; d < 32; d <<= 1) {
        float nv = __shfl_up(v, d, 32);
        if (lane >= d) v += nv;
    }
    if (lane == 31) wsumbuf[wid] = v;      // wave total
    __syncthreads();

    float stot = 0.0f, woff = 0.0f;
    #pragma unroll
    for (int w = 0; w < TPB / 32; ++w) {
        float s = wsumbuf[w];
        stot += s;
        if (w < wid) woff += s;
    }
    float excl = woff + (v - tsum);        // exclusive prefix for this thread
    float inv  = (stot != 0.0f) ? (1.0f / stot) : 0.0f;
    #pragma unroll
    for (int k = 0; k < EPT; ++k) lds_out[tid * EPT + k] = (excl + incl[k]) * inv;
    __syncthreads();                       // arrays complete; wsumbuf reusable
    return stot;
}

// ---------------- Kernel 3: one block per trace ----------------
// Exact decomposition of the merged-quantile integral:
//  * u-element i with q=u[i] > u[i-1]: delta = q - max(u[i-1], max{v < q});
//    indices iu=i, iv=min(#{v<q}, NT-1)  (ties: u precedes v, dup deltas are 0)
//  * v-element j with q=v[j] > v[j-1] and q not present in u:
//    delta = q - max(v[j-1], max{u <= q}); iu=min(lower_bound(u,q), NT-1), iv=j
// Matches reference searchsorted-left semantics (index choice only differs at
// zero-delta duplicate levels).
__global__ __launch_bounds__(TPB)
void ktrace(const float* __restrict__ x, const float* __restrict__ y,
            const float* __restrict__ mnp, float* __restrict__ costs) {
    __shared__ float su[NT];
    __shared__ float sv[NT];
    __shared__ float wsum[TPB / 32];
    __shared__ float part[TPB];

    int tid = threadIdx.x;
    int tr  = blockIdx.x;
    int b   = tr >> 11;          // tr / NPIX
    int p   = tr & (NPIX - 1);   // pixel index within (nr,nc) plane
    const float* xb = x + (size_t)b * NT * NPIX + p;
    const float* yb = y + (size_t)b * NT * NPIX + p;
    float mn = mnp[0];

    float lx[EPT], ly[EPT];
    #pragma unroll
    for (int k = 0; k < EPT; ++k) {
        int t = tid * EPT + k;
        lx[k] = xb[(size_t)t * NPIX] - mn;   // stride 8KB; L2-resident (128MB < 192MB L2)
        ly[k] = yb[(size_t)t * NPIX] - mn;
    }

    float stx = scan_normalize(su, lx, tid, wsum);
    float sty = scan_normalize(sv, ly, tid, wsum);
    bool valid = (stx != 0.0f) && (sty != 0.0f);

    float acc = 0.0f;
    // u-side levels
    #pragma unroll
    for (int k = 0; k < EPT; ++k) {
        int i = tid * EPT + k;
        float q = su[i];
        float uprev = (i > 0) ? su[i - 1] : 0.0f;
        if (q > uprev) {
            int lb = lower_bound_nt(sv, q);
            float vlt = (lb > 0) ? sv[lb - 1] : 0.0f;
            float pred = fmaxf(uprev, vlt);
            int iv = min(lb, NT - 1);
            float d = (float)(i - iv) * DTF;
            acc += (q - pred) * d * d;
        }
    }
    // v-side levels
    #pragma unroll
    for (int k = 0; k < EPT; ++k) {
        int j = tid * EPT + k;
        float q = sv[j];
        float vprev = (j > 0) ? sv[j - 1] : 0.0f;
        if (q > vprev) {
            int lb = lower_bound_nt(su, q);
            bool eq = (lb < NT) && (su[lb] == q);   // equal u level owns the delta
            if (!eq) {
                float ule = (lb > 0) ? su[lb - 1] : 0.0f;
                float pred = fmaxf(vprev, ule);
                int iu = min(lb, NT - 1);
                float d = (float)(iu - j) * DTF;
                acc += (q - pred) * d * d;
            }
        }
    }

    part[tid] = acc;
    __syncthreads();

    // WMMA-based block sum: A = ones(16x4), B = 64 partials -> D columns are exact
    // column sums; sum(c[0] over 32 lanes) = 2 * total regardless of B lane layout.
    if (tid < 32) {
        v8f c = {0.f, 0.f, 0.f, 0.f, 0.f, 0.f, 0.f, 0.f};
        v2f ones; ones[0] = 1.0f; ones[1] = 1.0f;
        #pragma unroll
        for (int ch = 0; ch < TPB / 64; ++ch) {
            v2f bv;
            bv[0] = part[ch * 64 + tid];
            bv[1] = part[ch * 64 + 32 + tid];
            c = __builtin_amdgcn_wmma_f32_16x16x4_f32(
                    false, ones, false, bv, (short)0, c, false, false);
        }
        float s = c[0];
        #pragma unroll
        for (int off = 16; off > 0; off >>= 1) s += __shfl_xor(s, off, 32);
        if (tid == 0) costs[tr] = valid ? 0.5f * s : 0.0f;
    }
}

// ---------------- Kernel 4: single-wave WMMA-accumulate final sum ----------------
__global__ __launch_bounds__(32)
void kfinal(const float* __restrict__ costs, float* __restrict__ out, int nvals) {
    int tid = threadIdx.x;
    v8f c = {0.f, 0.f, 0.f, 0.f, 0.f, 0.f, 0.f, 0.f};
    v2f ones; ones[0] = 1.0f; ones[1] = 1.0f;
    for (int base = 0; base < nvals; base += 64) {
        v2f bv;
        bv[0] = costs[base + tid];
        bv[1] = costs[base + 32 + tid];
        c = __builtin_amdgcn_wmma_f32_16x16x4_f32(
                false, ones, false, bv, (short)0, c, false, false);
    }
    float s = c[0];
    #pragma unroll
    for (int off = 16; off > 0; off >>= 1) s += __shfl_xor(s, off, 32);
    if (tid == 0) out[0] = 0.5f * s;
}

extern "C" void kernel_launch(void* const* d_in, const int* in_sizes, int n_in,
                              void* d_out, int out_size, void* d_ws, size_t ws_size,
                              hipStream_t stream) {
    const float* x = (const float*)d_in[0];
    const float* y = (const float*)d_in[1];
    float* out = (float*)d_out;

    float* ws    = (float*)d_ws;
    float* mn    = ws;                 // [0]
    float* part  = ws + 2;             // [2 .. 2+1024)
    float* costs = ws + 2 + 1024;      // [1026 .. 1026+8192)

    int n = in_sizes[0];               // 16,777,216 elements per array

    kmin_partial<<<1024, TPB, 0, stream>>>(x, y, part, n);
    kmin_final  <<<1,    TPB, 0, stream>>>(part, mn, 1024);
    ktrace      <<<NTRACE, TPB, 0, stream>>>(x, y, mn, costs);
    kfinal      <<<1, 32, 0, stream>>>(costs, out, NTRACE);
}